// HGRN_60851096649888
// MI455X (gfx1250) — compile-verified
//
#include <hip/hip_runtime.h>
#include <hip/hip_bf16.h>
#include <math.h>

typedef __attribute__((ext_vector_type(2))) float    v2f;
typedef __attribute__((ext_vector_type(8))) float    v8f;
typedef __attribute__((ext_vector_type(4))) unsigned v4u;
typedef __attribute__((ext_vector_type(4))) int      v4i;
typedef __attribute__((ext_vector_type(8))) int      v8i;

#define N_NODES   100000
#define N_EDGES   3200000
#define F_IN      512
#define HIDDEN    64
#define N_CLASSES 16
#define LEAKY     0.2f
#define LAMBDA_G  0.2f

#if __has_builtin(__builtin_amdgcn_tensor_load_to_lds) && __has_builtin(__builtin_amdgcn_s_wait_tensorcnt)
#define USE_TDM 1
#else
#define USE_TDM 0
#endif

// ---- monotone float <-> uint mapping so integer atomicMax == float max ----
__device__ __forceinline__ unsigned flipF(float f) {
  unsigned u = __float_as_uint(f);
  return (u & 0x80000000u) ? ~u : (u | 0x80000000u);
}
__device__ __forceinline__ float unflipF(unsigned u) {
  return __uint_as_float((u & 0x80000000u) ? (u ^ 0x80000000u) : ~u);
}

// =====================================================================
// Fused  Z = relu(X @ W0) @ W1   (fp32 WMMA, H staged in LDS only)
// block = 128 threads (4 waves), 64 rows per block.
// X tile staged via Tensor Data Mover (TDM); W0/W1 staged in a
// fragment-ready paired layout: float2(W[2p,n], W[2p+1,n]) at [p][n],
// so each WMMA B fragment is a single aligned ds_load_b64.
// =====================================================================
__global__ __launch_bounds__(128) void gemm_mlp_kernel(
    const float* __restrict__ X, const float* __restrict__ W0,
    const float* __restrict__ W1, float* __restrict__ Z) {
  __shared__ float sX[64 * 64];     // X tile (row-major; later reused for H)
  __shared__ float sW[64 * 64];     // W0 K-chunk, paired layout (32 pairs x 64 cols x 2)
  __shared__ float sW1[64 * 16];    // W1, paired layout (32 pairs x 16 cols x 2)

  const int tid  = threadIdx.x;
  const int wave = tid >> 5;
  const int lane = tid & 31;
  const int ln   = lane & 15;
  const int kh   = (lane >> 4) << 1;     // 0 or 2: K-half select (ISA A/B layout)
  const int ph   = lane >> 4;            // pair offset for the K-half
  const int rowBase = blockIdx.x * 64;

  v8f acc0 = {}, acc1 = {}, acc2 = {}, acc3 = {};

  for (int kc = 0; kc < F_IN; kc += 64) {
#if USE_TDM
    if (wave == 0) {
      // ---- TDM descriptor: 64x64 fp32 tile of X -> sX (row-major) ----
      unsigned long long gaddr =
          (unsigned long long)(uintptr_t)(X + (size_t)rowBase * F_IN + kc);
      unsigned ldsOff = (unsigned)(uintptr_t)(void*)sX;  // LDS aperture: low 32 bits = LDS byte offset
      unsigned dim0 = (unsigned)(F_IN - kc);      // elements remaining in a row
      unsigned dim1 = (unsigned)(N_NODES - rowBase); // rows remaining (OOB rows read as 0)
      v4u g0;
      g0.x = 1u;                                   // count=1, user descriptor
      g0.y = ldsOff;                               // lds_addr
      g0.z = (unsigned)(gaddr & 0xFFFFFFFFu);      // global_addr[31:0]
      g0.w = (unsigned)((gaddr >> 32) & 0x01FFFFFFu) | 0x80000000u; // addr[56:32] | type=2
      v8i g1;
      g1[0] = (int)(2u << 16);                     // workgroup_mask=0, data_size=4B
      g1[1] = (int)((dim0 & 0xFFFFu) << 16);       // tensor_dim0[15:0] @ bits63:48
      g1[2] = (int)((dim0 >> 16) & 0xFFFFu) | (int)((dim1 & 0xFFFFu) << 16); // dim0 hi | dim1 lo
      g1[3] = (int)((dim1 >> 16) & 0xFFFFu) | (int)(64u << 16);  // dim1 hi | tile_dim0=64
      g1[4] = 64;                                  // tile_dim1=64, tile_dim2=0
      g1[5] = F_IN;                                // tensor_dim0_stride = 512
      g1[6] = 0;
      g1[7] = 0;
      v4i zz4 = {0, 0, 0, 0};
      v8i zz8 = {0, 0, 0, 0, 0, 0, 0, 0};
      __builtin_amdgcn_tensor_load_to_lds(g0, g1, zz4, zz4, zz8, 0);
      __builtin_amdgcn_s_wait_tensorcnt(0);
    }
#else
    for (int i = 0; i < 8; ++i) {
      int slot = tid + i * 128;
      int r  = slot >> 4;
      int c4 = slot & 15;
      float4 xv = make_float4(0.f, 0.f, 0.f, 0.f);
      int grow = rowBase + r;
      if (grow < N_NODES)
        xv = ((const float4*)(X + (size_t)grow * F_IN + kc))[c4];
      ((float4*)sX)[slot] = xv;
    }
#endif
    // ---- W0 chunk -> paired layout: slot = p*32 + n2 holds
    //      {W0[2p,2n2], W0[2p+1,2n2], W0[2p,2n2+1], W0[2p+1,2n2+1]} ----
    for (int i = 0; i < 8; ++i) {
      int slot = tid + i * 128;       // 0..1023
      int p  = slot >> 5;             // K pair 0..31
      int n2 = slot & 31;             // column pair
      const float* g = W0 + (size_t)(kc + 2 * p) * HIDDEN + 2 * n2;
      float2 lo = *(const float2*)g;
      float2 hi = *(const float2*)(g + HIDDEN);
      ((float4*)sW)[slot] = make_float4(lo.x, hi.x, lo.y, hi.y);
    }
    __syncthreads();

    const float* aRow = sX + (wave * 16 + ln) * 64;
    for (int kk = 0; kk < 64; kk += 4) {
      v2f a = *(const v2f*)(aRow + kk + kh);
      const v2f* bp = (const v2f*)sW + ((kk >> 1) + ph) * 64 + ln;
      v2f b0 = bp[0], b1 = bp[16], b2 = bp[32], b3 = bp[48];
      acc0 = __builtin_amdgcn_wmma_f32_16x16x4_f32(false, a, false, b0, (short)0, acc0, false, false);
      acc1 = __builtin_amdgcn_wmma_f32_16x16x4_f32(false, a, false, b1, (short)0, acc1, false, false);
      acc2 = __builtin_amdgcn_wmma_f32_16x16x4_f32(false, a, false, b2, (short)0, acc2, false, false);
      acc3 = __builtin_amdgcn_wmma_f32_16x16x4_f32(false, a, false, b3, (short)0, acc3, false, false);
    }
    __syncthreads();
  }

  // ReLU + write H tile into sX (C/D layout: vgpr i -> M = i (+8 for hi lanes))
  {
    const int rbase = wave * 16 + (ph << 3);
    for (int i = 0; i < 8; ++i) {
      float* hr = sX + (rbase + i) * 64 + ln;
      hr[0]  = fmaxf(acc0[i], 0.f);
      hr[16] = fmaxf(acc1[i], 0.f);
      hr[32] = fmaxf(acc2[i], 0.f);
      hr[48] = fmaxf(acc3[i], 0.f);
    }
    // W1 -> paired layout: slot = p*8 + n2
    for (int i = 0; i < 2; ++i) {
      int slot = tid + i * 128;       // 0..255
      int p  = slot >> 3;
      int n2 = slot & 7;
      const float* g = W1 + (size_t)(2 * p) * N_CLASSES + 2 * n2;
      float2 lo = *(const float2*)g;
      float2 hi = *(const float2*)(g + N_CLASSES);
      ((float4*)sW1)[slot] = make_float4(lo.x, hi.x, lo.y, hi.y);
    }
  }
  __syncthreads();

  // Z = H @ W1  (16x16 out per wave, K=64)
  v8f zacc = {};
  {
    const float* aRow = sX + (wave * 16 + ln) * 64;
    for (int kk = 0; kk < 64; kk += 4) {
      v2f a = *(const v2f*)(aRow + kk + kh);
      v2f b = *((const v2f*)sW1 + ((kk >> 1) + ph) * 16 + ln);
      zacc = __builtin_amdgcn_wmma_f32_16x16x4_f32(false, a, false, b, (short)0, zacc, false, false);
    }
  }
  {
    const int rbase = wave * 16 + (ph << 3);
    for (int i = 0; i < 8; ++i) {
      int grow = rowBase + rbase + i;
      if (grow < N_NODES)
        Z[(size_t)grow * N_CLASSES + ln] = zacc[i];
    }
  }
}

// =====================================================================
// Per-hop sparse attention kernels
// =====================================================================
__global__ __launch_bounds__(256) void node_prep_kernel(
    const float* __restrict__ Zin, const float* __restrict__ a1,
    const float* __restrict__ a2, float* __restrict__ f1,
    float* __restrict__ f2, unsigned* __restrict__ m,
    float* __restrict__ s, float* __restrict__ outAcc) {
  int n = blockIdx.x * blockDim.x + threadIdx.x;
  if (n >= N_NODES) return;
  const float* z = Zin + (size_t)n * N_CLASSES;
  float d1 = 0.f, d2 = 0.f;
  for (int c = 0; c < N_CLASSES; ++c) {
    float zv = z[c];
    d1 += zv * a1[c];
    d2 += zv * a2[c];
  }
  f1[n] = d1;
  f2[n] = d2;
  m[n] = 0x007FFFFFu;  // flipF(-inf)
  s[n] = 0.f;
  float* o = outAcc + (size_t)n * N_CLASSES;
  for (int c = 0; c < N_CLASSES; ++c) o[c] = 0.f;
}

__global__ __launch_bounds__(256) void edge_pass1_kernel(
    const int* __restrict__ src, const int* __restrict__ dst,
    const float* __restrict__ f1, const float* __restrict__ f2,
    float* __restrict__ ebuf, unsigned* __restrict__ m) {
  int e = blockIdx.x * blockDim.x + threadIdx.x;
  if (e >= N_EDGES) return;
  int si = src[e], di = dst[e];
  float ev = f1[si] + f2[di];
  ev = ev > 0.f ? ev : LEAKY * ev;
  ebuf[e] = ev;
  atomicMax(m + si, flipF(ev));
}

__global__ __launch_bounds__(256) void edge_pass2_kernel(
    const int* __restrict__ src, float* __restrict__ ebuf,
    const unsigned* __restrict__ m, float* __restrict__ s) {
  int e = blockIdx.x * blockDim.x + threadIdx.x;
  if (e >= N_EDGES) return;
  int si = src[e];
  float ex = __expf(ebuf[e] - unflipF(m[si]));
  ebuf[e] = ex;
  atomicAdd(s + si, ex);
}

__global__ __launch_bounds__(256) void edge_pass3_kernel(
    const int* __restrict__ src, const int* __restrict__ dst,
    const float* __restrict__ ebuf, const float* __restrict__ s,
    const float* __restrict__ gcn, const float* __restrict__ Zin,
    float* __restrict__ outAcc) {
  long long t = (long long)blockIdx.x * blockDim.x + threadIdx.x;
  if (t >= (long long)N_EDGES * N_CLASSES) return;
  int e = (int)(t >> 4);
  int c = (int)(t & 15);
  int si = src[e], di = dst[e];
  float soft = ebuf[e] / s[si];
  float coef = gcn[e] * (1.0f - LAMBDA_G) + soft * LAMBDA_G;
  atomicAdd(outAcc + (size_t)si * N_CLASSES + c,
            coef * Zin[(size_t)di * N_CLASSES + c]);
}

// =====================================================================
// Final per-node attention over the 3 hop representations
// =====================================================================
__global__ __launch_bounds__(256) void final_attn_kernel(
    const float* __restrict__ P0, const float* __restrict__ P1,
    const float* __restrict__ P2, const float* __restrict__ Wattn,
    float* __restrict__ out) {
  int n = blockIdx.x * blockDim.x + threadIdx.x;
  if (n >= N_NODES) return;
  float p0[N_CLASSES], p1[N_CLASSES], p2[N_CLASSES], w[N_CLASSES];
  for (int c = 0; c < N_CLASSES; ++c) {
    p0[c] = P0[(size_t)n * N_CLASSES + c];
    p1[c] = P1[(size_t)n * N_CLASSES + c];
    p2[c] = P2[(size_t)n * N_CLASSES + c];
    w[c]  = Wattn[(size_t)n * N_CLASSES + c];
  }
  float a0 = 0.f, a1v = 0.f, a2v = 0.f;
  for (int c = 0; c < N_CLASSES; ++c) {
    a0  += p0[c] * w[c];
    a1v += p1[c] * w[c];
    a2v += p2[c] * w[c];
  }
  float mx = fmaxf(a0, fmaxf(a1v, a2v));
  float e0 = __expf(a0 - mx), e1 = __expf(a1v - mx), e2 = __expf(a2v - mx);
  float inv = 1.0f / (e0 + e1 + e2);
  e0 *= inv; e1 *= inv; e2 *= inv;
  for (int c = 0; c < N_CLASSES; ++c)
    out[(size_t)n * N_CLASSES + c] = p0[c] * e0 + p1[c] * e1 + p2[c] * e2;
}

// =====================================================================
extern "C" void kernel_launch(void* const* d_in, const int* in_sizes, int n_in,
                              void* d_out, int out_size, void* d_ws, size_t ws_size,
                              hipStream_t stream) {
  const float* X     = (const float*)d_in[0];
  const float* W0    = (const float*)d_in[1];
  const float* W1    = (const float*)d_in[2];
  const float* a1    = (const float*)d_in[3];
  const float* a2    = (const float*)d_in[4];
  const float* Wattn = (const float*)d_in[5];
  const float* gcn   = (const float*)d_in[6];
  const int*   ei    = (const int*)d_in[7];
  const int* src = ei;
  const int* dst = ei + N_EDGES;

  char* ws = (char*)d_ws;
  float* Z    = (float*)ws;            ws += (size_t)N_NODES * N_CLASSES * 4;
  float* h1   = (float*)ws;            ws += (size_t)N_NODES * N_CLASSES * 4;
  float* h2   = (float*)ws;            ws += (size_t)N_NODES * N_CLASSES * 4;
  float* f1   = (float*)ws;            ws += (size_t)N_NODES * 4;
  float* f2   = (float*)ws;            ws += (size_t)N_NODES * 4;
  unsigned* m = (unsigned*)ws;         ws += (size_t)N_NODES * 4;
  float* s    = (float*)ws;            ws += (size_t)N_NODES * 4;
  float* ebuf = (float*)ws;            ws += (size_t)N_EDGES * 4;

  // 1) fused MLP: Z = relu(X@W0)@W1
  gemm_mlp_kernel<<<(N_NODES + 63) / 64, 128, 0, stream>>>(X, W0, W1, Z);

  // 2) two sparse-attention hops
  const int nodeBlocks = (N_NODES + 255) / 256;
  const int edgeBlocks = (N_EDGES + 255) / 256;
  const int ecBlocks   = (int)(((long long)N_EDGES * N_CLASSES + 255) / 256);
  const float* hopIn = Z;
  float* hopOuts[2] = {h1, h2};
  for (int h = 0; h < 2; ++h) {
    float* hopOut = hopOuts[h];
    node_prep_kernel<<<nodeBlocks, 256, 0, stream>>>(hopIn, a1, a2, f1, f2, m, s, hopOut);
    edge_pass1_kernel<<<edgeBlocks, 256, 0, stream>>>(src, dst, f1, f2, ebuf, m);
    edge_pass2_kernel<<<edgeBlocks, 256, 0, stream>>>(src, ebuf, m, s);
    edge_pass3_kernel<<<ecBlocks, 256, 0, stream>>>(src, dst, ebuf, s, gcn, hopIn, hopOut);
    hopIn = hopOut;
  }

  // 3) per-node hop attention
  final_attn_kernel<<<nodeBlocks, 256, 0, stream>>>(Z, h1, h2, Wattn, (float*)d_out);
}